// BERT_9070970929347
// MI455X (gfx1250) — compile-verified
//
#include <hip/hip_runtime.h>
#include <hip/hip_bf16.h>

typedef __attribute__((ext_vector_type(2))) float v2f;
typedef __attribute__((ext_vector_type(8))) float v8f;

#define BSZ   4096
#define SSZ   128
#define DSZ   9
#define DP    12      // D padded to 3 k-chunks of 4
#define VSZ   5
#define NBLK  4
#define NW    4       // waves per workgroup (128 threads, wave32)
#define SSTR  130     // scores row stride in floats (bank-conflict padding)

__global__ __launch_bounds__(128)
void bert_fused_kernel(const int* __restrict__ tokens,
                       const float* __restrict__ emb,
                       const float* __restrict__ Wq, const float* __restrict__ bq,
                       const float* __restrict__ Wk, const float* __restrict__ bk,
                       const float* __restrict__ Wv, const float* __restrict__ bv,
                       const float* __restrict__ Wout, const float* __restrict__ bout,
                       float* __restrict__ out)
{
    __shared__ float xb[SSZ * DP];
    __shared__ float Qb[SSZ * DP];
    __shared__ float Kb[SSZ * DP];
    __shared__ float Vb[SSZ * DP];
    __shared__ float sc[NW][16 * SSTR];

    const int b    = blockIdx.x;
    const int tid  = threadIdx.x;
    const int wave = tid >> 5;
    const int lane = tid & 31;
    const int l    = lane & 15;   // lane % 16
    const int kh   = lane >> 4;   // half-wave select (k/row-half split)

    // ---- x = emb[tokens] + positional encoding, zero-padded to DP cols ----
    for (int idx = tid; idx < SSZ * DP; idx += blockDim.x) {
        int s = idx / DP, d = idx % DP;
        float v = 0.f;
        if (d < DSZ) {
            int tok   = tokens[b * SSZ + s];
            float frq = __powf(10000.f, (2.f * (float)(d >> 1)) / (float)DSZ);
            float ang = (float)s / frq;
            float pe  = (d & 1) ? __cosf(ang) : __sinf(ang);
            v = emb[tok * DSZ + d] + pe;
        }
        xb[idx] = v;
    }
    __syncthreads();

    const float scale = 1.f / 3.f;  // 1/sqrt(D), D=9; folded into Q

    for (int blk = 0; blk < NBLK; ++blk) {
        const float* wq = Wq + blk * DSZ * DSZ;  const float* bqv = bq + blk * DSZ;
        const float* wk = Wk + blk * DSZ * DSZ;  const float* bkv = bk + blk * DSZ;
        const float* wv = Wv + blk * DSZ * DSZ;  const float* bvv = bv + blk * DSZ;

        // ---- Q/K/V projections: y = x @ W^T + b (9x9, scalar FMAs) ----
        // Q is pre-scaled by 1/sqrt(D): (Q*s)@K^T == (Q@K^T)*s
        for (int idx = tid; idx < SSZ * DP; idx += blockDim.x) {
            int s = idx / DP, e = idx % DP;
            float q = 0.f, k = 0.f, vv = 0.f;
            if (e < DSZ) {
                q = bqv[e]; k = bkv[e]; vv = bvv[e];
                #pragma unroll
                for (int d = 0; d < DSZ; ++d) {
                    float xv = xb[s * DP + d];
                    q  = fmaf(xv, wq[e * DSZ + d], q);
                    k  = fmaf(xv, wk[e * DSZ + d], k);
                    vv = fmaf(xv, wv[e * DSZ + d], vv);
                }
                q *= scale;
            }
            Qb[idx] = q; Kb[idx] = k; Vb[idx] = vv;
        }
        __syncthreads();

        float* scw = &sc[wave][0];

        for (int mt = wave; mt < 8; mt += NW) {   // 16-row output tile per wave
            const int arow = mt * 16 + l;

            // ---- scores tile: (Q/sqrtD) @ K^T via V_WMMA_F32_16X16X4_F32 ----
            for (int nt = 0; nt < 8; ++nt) {
                v8f acc = {};
                #pragma unroll
                for (int kc = 0; kc < 3; ++kc) {
                    // A 16x4 layout: lanes 0-15 -> {K=k0,k0+1}, lanes 16-31 -> {k0+2,k0+3}
                    int k0 = kc * 4 + 2 * kh;
                    v2f a, bb;
                    a.x  = Qb[arow * DP + k0];
                    a.y  = Qb[arow * DP + k0 + 1];
                    // B 4x16 = K^T: B[k,n] = K[n,k]
                    bb.x = Kb[(nt * 16 + l) * DP + k0];
                    bb.y = Kb[(nt * 16 + l) * DP + k0 + 1];
                    acc = __builtin_amdgcn_wmma_f32_16x16x4_f32(
                              false, a, false, bb, (short)0, acc, false, false);
                }
                // C/D layout: lanes 0-15 rows 0-7, lanes 16-31 rows 8-15
                #pragma unroll
                for (int r = 0; r < 8; ++r)
                    scw[(kh * 8 + r) * SSTR + nt * 16 + l] = acc[r];
            }
            __builtin_amdgcn_wave_barrier();

            // ---- wave-synchronous softmax (unnormalized exp; inv kept in reg) ----
            float inv;
            {
                float* row = &scw[l * SSTR];   // lane pair (l, l+16) shares row l
                int j0 = kh * 64;
                float m = -__builtin_inff();
                for (int j = j0; j < j0 + 64; ++j) m = fmaxf(m, row[j]);
                m = fmaxf(m, __shfl_xor(m, 16, 32));
                float ssum = 0.f;
                for (int j = j0; j < j0 + 64; ++j) {
                    float e = __expf(row[j] - m);
                    row[j] = e;
                    ssum += e;
                }
                ssum += __shfl_xor(ssum, 16, 32);
                inv = 1.f / ssum;              // valid in both lanes of the pair
            }
            __builtin_amdgcn_wave_barrier();

            // ---- x_new tile = (exp @ V) * inv_row via WMMA fp32, K=128 ----
            v8f acc = {};
            for (int kc = 0; kc < 32; ++kc) {
                int k0 = kc * 4 + 2 * kh;
                v2f a, bb;
                a.x  = scw[l * SSTR + k0];
                a.y  = scw[l * SSTR + k0 + 1];
                bb.x = Vb[(k0)     * DP + l];
                bb.y = Vb[(k0 + 1) * DP + l];
                acc = __builtin_amdgcn_wmma_f32_16x16x4_f32(
                          false, a, false, bb, (short)0, acc, false, false);
            }
            // normalize per row: row (kh*8+r)'s inv is held by lane (kh*8+r)
            #pragma unroll
            for (int r = 0; r < 8; ++r)
                acc[r] *= __shfl(inv, kh * 8 + r, 32);
            if (l < DP) {
                #pragma unroll
                for (int r = 0; r < 8; ++r)
                    xb[(mt * 16 + kh * 8 + r) * DP + l] = acc[r];
            }
        }
        __syncthreads();
    }

    // ---- logits + log_softmax (V=5) ----
    if (tid < SSZ) {
        int s = tid;
        float lg[VSZ];
        float m = -__builtin_inff();
        #pragma unroll
        for (int v = 0; v < VSZ; ++v) {
            float a = bout[v];
            #pragma unroll
            for (int d = 0; d < DSZ; ++d)
                a = fmaf(xb[s * DP + d], Wout[v * DSZ + d], a);
            lg[v] = a;
            m = fmaxf(m, a);
        }
        float ssum = 0.f;
        #pragma unroll
        for (int v = 0; v < VSZ; ++v) ssum += __expf(lg[v] - m);
        float lse = m + __logf(ssum);
        size_t base = ((size_t)b * SSZ + s) * VSZ;
        #pragma unroll
        for (int v = 0; v < VSZ; ++v) out[base + v] = lg[v] - lse;
    }
}

extern "C" void kernel_launch(void* const* d_in, const int* in_sizes, int n_in,
                              void* d_out, int out_size, void* d_ws, size_t ws_size,
                              hipStream_t stream) {
    const int*   tokens = (const int*)  d_in[0];
    const float* emb    = (const float*)d_in[1];
    const float* Wq     = (const float*)d_in[2];
    const float* bq     = (const float*)d_in[3];
    const float* Wk     = (const float*)d_in[4];
    const float* bk     = (const float*)d_in[5];
    const float* Wv     = (const float*)d_in[6];
    const float* bv     = (const float*)d_in[7];
    const float* Wout   = (const float*)d_in[8];
    const float* bout   = (const float*)d_in[9];
    float* out = (float*)d_out;

    bert_fused_kernel<<<BSZ, 128, 0, stream>>>(tokens, emb, Wq, bq, Wk, bk,
                                               Wv, bv, Wout, bout, out);
}